// BoxConv2d_47880295416171
// MI455X (gfx1250) — compile-verified
//
#include <hip/hip_runtime.h>
#include <hip/hip_bf16.h>

// ---------------------------------------------------------------------------
// BoxConv2d for MI455X (gfx1250, wave32)
//   B=8, C=32, F=4, H=W=128, REPARAM = 1024
// Pass 1: per-(b,c) 2-D integral image via WMMA triangular-matmul scan in LDS.
//         (prefix-sum = multiply by triangular ones-matrix; 16x16 tiles,
//          V_WMMA_F32_16X16X4_F32 with K split in 4 chunks, carry via
//          ds_bpermute lane broadcast)
// Pass 2: bilinear 4-corner box evaluation from LDS-staged ii rows.
//         Output streamed with non-temporal stores so the integral image
//         stays resident in the 192 MB L2.
// ---------------------------------------------------------------------------

typedef __attribute__((ext_vector_type(2))) float v2f;
typedef __attribute__((ext_vector_type(8))) float v8f;

#define LDSP      132          // LDS row pitch (floats): 16B-aligned rows, bank stride 4
#define II_PITCH  132          // workspace ii row pitch (floats)
#define II_ROWS   129          // ii rows (H+1)
#define II_IMG    (II_PITCH * II_ROWS)   // floats per (b,c) ii image
#define RHW       1024.0f      // REPARAM_H == REPARAM_W

__device__ __forceinline__ float bcast_lane(float v, int srcLane) {
    int r = __builtin_amdgcn_ds_bpermute(srcLane << 2, __builtin_bit_cast(int, v));
    return __builtin_bit_cast(float, r);
}

// ------------------------- Pass 1: integral image --------------------------
// grid = 256 blocks (one per (b,c) plane), 256 threads (8 waves).
__global__ void __launch_bounds__(256)
boxconv_integral_image(const float* __restrict__ input, float* __restrict__ ii) {
    __shared__ float smem[128 * LDSP];    // 67,584 B

    const int bc   = blockIdx.x;          // b*32 + c
    const int tid  = threadIdx.x;
    const int lane = tid & 31;
    const int wv   = tid >> 5;            // wave id 0..7
    const int lo   = lane & 15;
    const int hi   = lane >> 4;           // 0 | 1

    // ---- load 128x128 plane into LDS (float4, fully coalesced) ----
    const float* src = input + (size_t)bc * (128 * 128);
    #pragma unroll
    for (int i = tid; i < 128 * 32; i += 256) {       // 32 float4 per row
        int r = i >> 5, seg = i & 31;
        float4 v = ((const float4*)(src + r * 128))[seg];
        *(float4*)&smem[r * LDSP + seg * 4] = v;
    }
    __syncthreads();

    // ---- constant upper-triangular B chunks: U[k][n] = (k <= n) ----
    v2f Bu[4];
    #pragma unroll
    for (int k = 0; k < 4; ++k) {
        int K0 = 4 * k + 2 * hi;
        Bu[k].x = (K0     <= lo) ? 1.0f : 0.0f;
        Bu[k].y = (K0 + 1 <= lo) ? 1.0f : 0.0f;
    }
    const int carrySrc = (lane & 16) | 15;            // lane 15 / 31 per half

    // ---- row pass: each wave scans rows [16*wv, 16*wv+16) along W ----
    {
        const int r0 = wv * 16;
        v8f carry = {};
        for (int q = 0; q < 8; ++q) {
            v8f acc = carry;
            #pragma unroll
            for (int k = 0; k < 4; ++k) {
                int base = (r0 + lo) * LDSP + q * 16 + 4 * k + 2 * hi;
                v2f a;  a.x = smem[base];  a.y = smem[base + 1];
                acc = __builtin_amdgcn_wmma_f32_16x16x4_f32(
                        false, a, false, Bu[k], (short)0, acc, false, false);
            }
            #pragma unroll
            for (int v = 0; v < 8; ++v)
                smem[(r0 + v + 8 * hi) * LDSP + q * 16 + lo] = acc[v];
            #pragma unroll
            for (int v = 0; v < 8; ++v)
                carry[v] = bcast_lane(acc[v], carrySrc);
        }
    }
    __syncthreads();

    // ---- column pass: wave scans cols [16*wv, ...) along H, writes transposed
    //      in place so LDS ends up holding the natural-layout integral image --
    {
        const int c0 = wv * 16;
        v8f carry = {};
        for (int p = 0; p < 8; ++p) {
            v8f acc = carry;
            #pragma unroll
            for (int k = 0; k < 4; ++k) {
                int row = p * 16 + 4 * k + 2 * hi;
                v2f a;
                a.x = smem[row * LDSP + c0 + lo];
                a.y = smem[(row + 1) * LDSP + c0 + lo];
                acc = __builtin_amdgcn_wmma_f32_16x16x4_f32(
                        false, a, false, Bu[k], (short)0, acc, false, false);
            }
            #pragma unroll
            for (int v = 0; v < 8; ++v)
                smem[(p * 16 + lo) * LDSP + c0 + v + 8 * hi] = acc[v];
            #pragma unroll
            for (int v = 0; v < 8; ++v)
                carry[v] = bcast_lane(acc[v], carrySrc);
        }
    }
    __syncthreads();

    // ---- coalesced store of padded (129 x 129) ii into workspace ----
    // Regular (RT) stores: pass 2 re-reads ii, keep it L2-resident.
    float* dst = ii + (size_t)bc * II_IMG;
    for (int i = tid; i < II_IMG; i += 256) {
        int r = i / II_PITCH, y = i - r * II_PITCH;
        float val = (r == 0 || y == 0 || y > 128)
                  ? 0.0f
                  : smem[(r - 1) * LDSP + (y - 1)];
        dst[i] = val;
    }
}

// ------------------------- Pass 2: box evaluation --------------------------
// grid = (H=128, C*F=128), 256 threads. Block handles all 8 batches x 128 w.
__global__ void __launch_bounds__(256)
boxconv_eval(const float* __restrict__ ii,
             const float* __restrict__ xmn, const float* __restrict__ xmx,
             const float* __restrict__ ymn, const float* __restrict__ ymx,
             float* __restrict__ out) {
    const int h  = blockIdx.x;
    const int cf = blockIdx.y;
    const int c  = cf >> 2;
    const int f  = cf & 3;
    const int pi = c * 4 + f;

    const float xmin = xmn[pi] * RHW, xmax = xmx[pi] * RHW;
    const float ymin = ymn[pi] * RHW, ymax = ymx[pi] * RHW;

    // x (row) interpolation parameters: shared by the whole block
    float xs_t = fminf(fmaxf((float)h + xmin,        0.0f), 128.0f);
    float xs_b = fminf(fmaxf((float)h + xmax + 1.0f, 0.0f), 128.0f);
    float x0tf = fminf(floorf(xs_t), 127.0f);
    float x0bf = fminf(floorf(xs_b), 127.0f);
    float fxt  = xs_t - x0tf;
    float fxb  = xs_b - x0bf;
    int   x0t  = (int)x0tf;
    int   x0b  = (int)x0bf;
    int rowIdx[4] = { x0t, x0t + 1, x0b, x0b + 1 };

    // stage the 4 needed ii rows for all 8 batches into LDS (coalesced L2 hits)
    __shared__ float rows[8][4][II_PITCH];            // 16,896 B
    for (int i = threadIdx.x; i < 8 * 4 * II_PITCH; i += 256) {
        int b  = i / (4 * II_PITCH);
        int rr = (i / II_PITCH) & 3;
        int y  = i - (i / II_PITCH) * II_PITCH;
        float v = 0.0f;
        if (y < 129)
            v = ii[(size_t)(b * 32 + c) * II_IMG + rowIdx[rr] * II_PITCH + y];
        ((float*)rows)[i] = v;
    }
    __syncthreads();

    // per-thread y (column) parameters (w fixed per thread)
    const int w  = threadIdx.x & 127;
    const int b0 = threadIdx.x >> 7;                  // 0 | 1
    float ys_l = fminf(fmaxf((float)w + ymin,        0.0f), 128.0f);
    float ys_r = fminf(fmaxf((float)w + ymax + 1.0f, 0.0f), 128.0f);
    float y0lf = fminf(floorf(ys_l), 127.0f);
    float y0rf = fminf(floorf(ys_r), 127.0f);
    float fyl  = ys_l - y0lf;
    float fyr  = ys_r - y0rf;
    int   y0l  = (int)y0lf;
    int   y0r  = (int)y0rf;

    const float inv_area = 1.0f / ((xmax - xmin + 1.0f) * (ymax - ymin + 1.0f));

    auto samp = [&](const float* r0, const float* r1, float fx, int y0, float fy) {
        float a0 = (1.0f - fx) * r0[y0]     + fx * r1[y0];
        float a1 = (1.0f - fx) * r0[y0 + 1] + fx * r1[y0 + 1];
        return (1.0f - fy) * a0 + fy * a1;
    };

    #pragma unroll
    for (int j = 0; j < 4; ++j) {
        int b = b0 * 4 + j;
        float Bt_l = samp(rows[b][0], rows[b][1], fxt, y0l, fyl);
        float Bt_r = samp(rows[b][0], rows[b][1], fxt, y0r, fyr);
        float Bb_l = samp(rows[b][2], rows[b][3], fxb, y0l, fyl);
        float Bb_r = samp(rows[b][2], rows[b][3], fxb, y0r, fyr);
        float o = (Bb_r - Bb_l - Bt_r + Bt_l) * inv_area;
        // 64 MB write-once stream: non-temporal, bypass L2 residency so the
        // integral image stays cached for the remaining blocks.
        __builtin_nontemporal_store(
            o, &out[((size_t)((b * 32 + c) * 4 + f)) * (128 * 128) + h * 128 + w]);
    }
}

// ---------------------------------------------------------------------------
extern "C" void kernel_launch(void* const* d_in, const int* in_sizes, int n_in,
                              void* d_out, int out_size, void* d_ws, size_t ws_size,
                              hipStream_t stream) {
    (void)in_sizes; (void)n_in; (void)out_size; (void)ws_size;
    const float* input = (const float*)d_in[0];
    const float* x_min = (const float*)d_in[1];
    const float* x_max = (const float*)d_in[2];
    const float* y_min = (const float*)d_in[3];
    const float* y_max = (const float*)d_in[4];
    float* out = (float*)d_out;
    float* ii  = (float*)d_ws;     // 256 * 129*132 * 4 B ~= 16.6 MB

    boxconv_integral_image<<<dim3(8 * 32), dim3(256), 0, stream>>>(input, ii);
    boxconv_eval<<<dim3(128, 128), dim3(256), 0, stream>>>(
        ii, x_min, x_max, y_min, y_max, out);
}